// GrassmannianLanguageModel_51402168598969
// MI455X (gfx1250) — compile-verified
//
#include <hip/hip_runtime.h>
#include <hip/hip_bf16.h>

// Problem dims (from reference)
constexpr int VOCAB   = 50257;
constexpr int D_MODEL = 1024;
constexpr int NDIM    = 512;   // n
constexpr int KDIM    = 64;    // k
constexpr int BATCH   = 4;
constexpr int SEQ     = 2048;

constexpr int VTILES  = (VOCAB + 15) / 16;   // 3142
constexpr int WAVES   = 8;                   // waves (vtiles) per block in score kernel
constexpr int NPASS_ROWS = 256;              // U rows staged in LDS per pass (64 KB)

typedef __attribute__((ext_vector_type(2))) float    v2f;
typedef __attribute__((ext_vector_type(8))) float    v8f;
typedef __attribute__((ext_vector_type(4))) unsigned v4ui;
typedef __attribute__((ext_vector_type(8))) int      v8i;
typedef __attribute__((ext_vector_type(4))) int      v4i;

// ---------------------------------------------------------------------------
// Kernel 1: ctx[b][d] = mean_l inp_emb[ids[b][l]][d]
// grid (D_MODEL/256, B), block 256
// ---------------------------------------------------------------------------
__global__ void ctx_mean_kernel(const int* __restrict__ ids,
                                const float* __restrict__ emb,
                                float* __restrict__ ctx) {
    const int d = blockIdx.x * blockDim.x + threadIdx.x;
    const int b = blockIdx.y;
    const int* row = ids + (size_t)b * SEQ;
    float acc = 0.0f;
    #pragma unroll 4
    for (int l = 0; l < SEQ; ++l) {
        const int id = row[l];
        acc += emb[(size_t)id * D_MODEL + d];
    }
    ctx[(size_t)b * D_MODEL + d] = acc * (1.0f / (float)SEQ);
}

// ---------------------------------------------------------------------------
// Kernel 2: U[b][j] = dot(ctx[b], W[j]) + bias[j]     (j = 0..N*K-1)
// One wave per W row; 4 batch accumulators so W is streamed exactly once.
// ---------------------------------------------------------------------------
__global__ void u_proj_kernel(const float* __restrict__ ctx,
                              const float* __restrict__ W,
                              const float* __restrict__ bias,
                              float* __restrict__ U) {
    const int wid  = (blockIdx.x * blockDim.x + threadIdx.x) >> 5;  // W row
    const int lane = threadIdx.x & 31;
    if (wid >= NDIM * KDIM) return;  // whole wave exits together
    const float* wrow = W + (size_t)wid * D_MODEL;
    float a0 = 0.0f, a1 = 0.0f, a2 = 0.0f, a3 = 0.0f;
    for (int d = lane; d < D_MODEL; d += 32) {
        const float wv = wrow[d];
        a0 += wv * ctx[0 * D_MODEL + d];
        a1 += wv * ctx[1 * D_MODEL + d];
        a2 += wv * ctx[2 * D_MODEL + d];
        a3 += wv * ctx[3 * D_MODEL + d];
    }
    #pragma unroll
    for (int off = 16; off > 0; off >>= 1) {
        a0 += __shfl_down(a0, off, 32);
        a1 += __shfl_down(a1, off, 32);
        a2 += __shfl_down(a2, off, 32);
        a3 += __shfl_down(a3, off, 32);
    }
    if (lane == 0) {
        const float bb = bias[wid];
        U[0 * (size_t)(NDIM * KDIM) + wid] = a0 + bb;
        U[1 * (size_t)(NDIM * KDIM) + wid] = a1 + bb;
        U[2 * (size_t)(NDIM * KDIM) + wid] = a2 + bb;
        U[3 * (size_t)(NDIM * KDIM) + wid] = a3 + bb;
    }
}

// ---------------------------------------------------------------------------
// Kernel 3: logits[b][v] = sum_k ( sum_n geom[v][n] * U[b][n][k] )^2
// One 16-row vocab tile per wave; U_b staged through LDS (TDM DMA when
// available) in two 256-row passes; fp32 WMMA (V_WMMA_F32_16X16X4_F32)
// accumulation over n.
// grid (ceil(VTILES/WAVES), B), block 32*WAVES
// ---------------------------------------------------------------------------
__global__ void __launch_bounds__(32 * WAVES)
score_kernel(const float* __restrict__ geom,
             const float* __restrict__ U,
             float* __restrict__ out) {
    __shared__ float sU[NPASS_ROWS * KDIM];  // 64 KB

    const int b     = blockIdx.y;
    const int wave  = threadIdx.x >> 5;
    const int lane  = threadIdx.x & 31;
    const int r     = lane & 15;       // row within tile / column within D
    const int khalf = lane >> 4;       // which K-pair of the WMMA fragment

    int vtile = blockIdx.x * WAVES + wave;
    if (vtile >= VTILES) vtile = VTILES - 1;          // keep EXEC all-ones
    int vrow = vtile * 16 + r;
    if (vrow >= VOCAB) vrow = VOCAB - 1;              // clamp ragged tail

    const float* arow = geom + (size_t)vrow * NDIM;   // this lane's A row
    const float* Ub   = U + (size_t)b * NDIM * KDIM;

    v8f acc0 = {}, acc1 = {}, acc2 = {}, acc3 = {};

    for (int pass = 0; pass < 2; ++pass) {
        // ---- Stage 256 U rows (64 KB) into LDS ----------------------------
#if defined(__gfx1250__) && __has_builtin(__builtin_amdgcn_tensor_load_to_lds)
        if (wave == 0) {
            // Tensor DMA descriptor (ISA 8.3/8.4): 2-D tile 64 x 256 f32,
            // contiguous in both global and LDS.
            const unsigned long long ga =
                (unsigned long long)(const void*)(Ub + (size_t)pass * NPASS_ROWS * KDIM);
            const unsigned lds = (unsigned)(__SIZE_TYPE__)&sU[0];
            v4ui g0 = { 1u,                                   // count=1 (valid user D#)
                        lds,                                  // lds_addr [63:32]
                        (unsigned)(ga & 0xFFFFFFFFu),         // global_addr lo
                        ((unsigned)((ga >> 32) & 0x01FFFFFFu))// global_addr hi [120:96]
                          | (2u << 30) };                     // type=2 ("image")
            v8i g1 = { (int)(2u << 16),                       // data_size=2 -> 4 bytes
                       (int)((unsigned)KDIM << 16),           // tensor_dim0 lo16
                       (int)((unsigned)NPASS_ROWS << 16),     // tensor_dim1 lo16
                       (int)((unsigned)KDIM << 16),           // tile_dim0
                       (int)NPASS_ROWS,                       // tile_dim1
                       (int)KDIM,                             // tensor_dim0_stride lo32
                       0, 0 };
            v4i g2 = { 0, 0, 0, 0 };
            v4i g3 = { 0, 0, 0, 0 };
            v8i g4 = { 0, 0, 0, 0, 0, 0, 0, 0 };              // unused (2-D tile)
            __builtin_amdgcn_tensor_load_to_lds(g0, g1, g2, g3, g4, 0);
            __builtin_amdgcn_s_wait_tensorcnt(0);             // DMA complete
        }
#else
        {
            const float4* src = (const float4*)(Ub + (size_t)pass * NPASS_ROWS * KDIM);
            float4* dst = (float4*)sU;
            for (int i = threadIdx.x; i < NPASS_ROWS * KDIM / 4; i += blockDim.x)
                dst[i] = src[i];
        }
#endif
        __syncthreads();

        const float* gp = arow + pass * NPASS_ROWS;   // geom cols for this pass
        #pragma unroll 2
        for (int nb = 0; nb < NPASS_ROWS / 4; ++nb) {
            const int nl = nb * 4 + khalf * 2;        // local n of this lane's K-pair
            // A fragment: geom[vrow][n], geom[vrow][n+1] (contiguous, b64 load)
            const v2f a = *(const v2f*)(gp + nl);
            __builtin_prefetch(gp + nl + 64, 0, 3);   // WGP-scope stream-ahead

            // B fragments from LDS: U[n][kcol], U[n+1][kcol] for 4 k-tiles
            const float* l0 = sU + nl * KDIM + r;
            v2f b0; b0.x = l0[0];  b0.y = l0[KDIM + 0];
            v2f b1; b1.x = l0[16]; b1.y = l0[KDIM + 16];
            v2f b2; b2.x = l0[32]; b2.y = l0[KDIM + 32];
            v2f b3; b3.x = l0[48]; b3.y = l0[KDIM + 48];

            acc0 = __builtin_amdgcn_wmma_f32_16x16x4_f32(false, a, false, b0,
                                                         (short)0, acc0, false, false);
            acc1 = __builtin_amdgcn_wmma_f32_16x16x4_f32(false, a, false, b1,
                                                         (short)0, acc1, false, false);
            acc2 = __builtin_amdgcn_wmma_f32_16x16x4_f32(false, a, false, b2,
                                                         (short)0, acc2, false, false);
            acc3 = __builtin_amdgcn_wmma_f32_16x16x4_f32(false, a, false, b3,
                                                         (short)0, acc3, false, false);
        }
        __syncthreads();  // protect LDS before next pass overwrites it
    }

    // Per-lane partial: row (khalf*8 + j), col r holds acc*[j]; square and
    // reduce over the 16 columns (XOR butterfly stays within each half-wave).
    float p[8];
    #pragma unroll
    for (int j = 0; j < 8; ++j)
        p[j] = acc0[j] * acc0[j] + acc1[j] * acc1[j] +
               acc2[j] * acc2[j] + acc3[j] * acc3[j];
    #pragma unroll
    for (int m = 1; m <= 8; m <<= 1) {
        #pragma unroll
        for (int j = 0; j < 8; ++j)
            p[j] += __shfl_xor(p[j], m, 32);
    }

    // Lane with (lane&15)==j in each half writes row khalf*8 + j.
    const int vbase = vtile * 16 + khalf * 8;
    float* outb = out + (size_t)b * VOCAB;
    #pragma unroll
    for (int j = 0; j < 8; ++j) {
        if (r == j) {
            const int v = vbase + j;
            if (v < VOCAB) outb[v] = p[j];
        }
    }
}

// ---------------------------------------------------------------------------
extern "C" void kernel_launch(void* const* d_in, const int* in_sizes, int n_in,
                              void* d_out, int out_size, void* d_ws, size_t ws_size,
                              hipStream_t stream) {
    const int*   ids  = (const int*)d_in[0];
    const float* emb  = (const float*)d_in[1];
    const float* W    = (const float*)d_in[2];
    const float* bias = (const float*)d_in[3];
    const float* geom = (const float*)d_in[4];
    float* out = (float*)d_out;

    // Workspace layout: ctx (B*D_MODEL) | U (B*N*K)
    float* ctx = (float*)d_ws;
    float* U   = ctx + (size_t)BATCH * D_MODEL;

    ctx_mean_kernel<<<dim3(D_MODEL / 256, BATCH), 256, 0, stream>>>(ids, emb, ctx);

    const int nwaves = NDIM * KDIM;                 // one wave per W row
    u_proj_kernel<<<(nwaves * 32) / 256, 256, 0, stream>>>(ctx, W, bias, U);

    const int vgroups = (VTILES + WAVES - 1) / WAVES;
    score_kernel<<<dim3(vgroups, BATCH), 32 * WAVES, 0, stream>>>(geom, U, out);
}